// BlockDiagonalConv2d_v2_89026082112076
// MI455X (gfx1250) — compile-verified
//
#include <hip/hip_runtime.h>

// Block-diagonal 3x3 conv (64 heads, 4->4 ch/head) on fp32 via V_WMMA_F32_16X16X4_F32.
//
// Roofline: x=512MiB + out=512MiB -> ~1.07 GB HBM -> ~46 us floor @ 23.3 TB/s.
// 9.7 GFLOP useful; scalar fp32 VALU (36 FMA/out) would be issue/VALU-bound, so the
// MACs go to the matrix pipe: per input row, 3x WMMA f32 16x16x4 (K=ci, one per kw),
// A rows = 4*kh+co (12/16 used), C row-groups land in output rows h+1, h, h-1 and are
// merged by a rolling 2-row accumulator + one half-swap ds_swizzle per co.

typedef __attribute__((ext_vector_type(2))) float v2f;
typedef __attribute__((ext_vector_type(8))) float v8f;

#define NHEAD 64
#define HW    128
#define NCH   256
#define CHSTR (HW * HW)

// ds_swizzle SWAPX16 (group-of-32: xor=0x10, or=0, and=0x1f) -> 0x401F
#define SWAP16_IMM 0x401F

static __device__ __forceinline__ float swap16(float v) {
    return __uint_as_float((unsigned)__builtin_amdgcn_ds_swizzle(
        (int)__float_as_uint(v), SWAP16_IMM));
}

template <bool EDGE>
static __device__ __forceinline__ void conv_sweep(
    const float* __restrict__ p0,   // lane ptr into ci0 plane, row 0, at w0+n-1 (interior)
    const float* __restrict__ p1,   // lane ptr into ci1 plane
    float* __restrict__ oA,         // lane store ptr, pair (co 0|2), row 0
    float* __restrict__ oB,         // lane store ptr, pair (co 1|3), row 0
    const v2f a0, const v2f a1, const v2f a2,
    const float b0, const float b1, const float b2, const float b3,
    const bool lower,
    const int ok0, const int ok1, const int ok2,   // EDGE only: w-in-range per kw
    const int c0off, const int c1off, const int c2off) // EDGE only: clamped byte-ish offsets
{
    float accA[4], accB[4];
    const float bsc[4] = {b0, b1, b2, b3};

    // ---- row 0 (peeled: no store yet) ----
    v2f bm0, bm1, bm2;
    {
        if (EDGE) {
            bm0.x = ok0 ? p0[c0off] : 0.f;  bm0.y = ok0 ? p1[c0off] : 0.f;
            bm1.x = ok1 ? p0[c1off] : 0.f;  bm1.y = ok1 ? p1[c1off] : 0.f;
            bm2.x = ok2 ? p0[c2off] : 0.f;  bm2.y = ok2 ? p1[c2off] : 0.f;
        } else {
            bm0.x = p0[0]; bm0.y = p1[0];
            bm1.x = p0[1]; bm1.y = p1[1];
            bm2.x = p0[2]; bm2.y = p1[2];
        }
        v8f c = {};
        c = __builtin_amdgcn_wmma_f32_16x16x4_f32(false, a0, false, bm0, (short)0, c, false, false);
        c = __builtin_amdgcn_wmma_f32_16x16x4_f32(false, a1, false, bm1, (short)0, c, false, false);
        c = __builtin_amdgcn_wmma_f32_16x16x4_f32(false, a2, false, bm2, (short)0, c, false, false);
#pragma unroll
        for (int co = 0; co < 4; ++co) {
            accA[co] = bsc[co] + c[4 + co];      // bias + R1(row0); R0(row -1) = pad
            accB[co] = c[co] + bsc[co];          // R0(row1) + bias
        }
        p0 += HW; p1 += HW;
    }

    // ---- rows 1..127: compute C_h, finish & store row h-1 ----
    for (int h = 1; h < HW; ++h) {
        if (EDGE) {
            bm0.x = ok0 ? p0[c0off] : 0.f;  bm0.y = ok0 ? p1[c0off] : 0.f;
            bm1.x = ok1 ? p0[c1off] : 0.f;  bm1.y = ok1 ? p1[c1off] : 0.f;
            bm2.x = ok2 ? p0[c2off] : 0.f;  bm2.y = ok2 ? p1[c2off] : 0.f;
        } else {
            bm0.x = p0[0]; bm0.y = p1[0];
            bm1.x = p0[1]; bm1.y = p1[1];
            bm2.x = p0[2]; bm2.y = p1[2];
        }
        v8f c = {};
        c = __builtin_amdgcn_wmma_f32_16x16x4_f32(false, a0, false, bm0, (short)0, c, false, false);
        c = __builtin_amdgcn_wmma_f32_16x16x4_f32(false, a1, false, bm1, (short)0, c, false, false);
        c = __builtin_amdgcn_wmma_f32_16x16x4_f32(false, a2, false, bm2, (short)0, c, false, false);

        // Finish output row h-1 with full-width pair stores:
        //  lanes 0..15 : co=0 (pairA) / co=1 (pairB):  accA[co] + swap16(c[co])   (R2 down)
        //  lanes 16..31: co=2 (pairA) / co=3 (pairB):  swap16(accA[co]) + c[co]   (acc up, R2 local)
        {
            const float s0 = swap16(c[0]);
            const float u2 = swap16(accA[2]);
            const float vA = lower ? (accA[0] + s0) : (u2 + c[2]);
            const float s1 = swap16(c[1]);
            const float u3 = swap16(accA[3]);
            const float vB = lower ? (accA[1] + s1) : (u3 + c[3]);
            *oA = vA;
            *oB = vB;
            oA += HW; oB += HW;
        }

        // Advance rolling state (R1 finishes kh=1 for row h; R0 starts row h+1).
#pragma unroll
        for (int co = 0; co < 4; ++co) {
            accA[co] = accB[co] + c[4 + co];
            accB[co] = c[co] + bsc[co];
        }
        p0 += HW; p1 += HW;
    }

    // ---- row 127: R2 from row 128 is zero padding ----
    {
        const float vA = lower ? accA[0] : swap16(accA[2]);
        const float vB = lower ? accA[1] : swap16(accA[3]);
        // note: for upper lanes swap16(accA[c]) fetches the lower-lane accumulator;
        // the value the lower lane "sends up" is its own accA which is what we need.
        *oA = vA;
        *oB = vB;
    }
}

__global__ __launch_bounds__(256)
void bdconv_wmma_f32_kernel(const float* __restrict__ x,
                            const float* __restrict__ wts,
                            const float* __restrict__ bias,
                            float* __restrict__ out)
{
    const int lane = threadIdx.x & 31;
    const int wid  = blockIdx.x * 8 + (threadIdx.x >> 5);   // 0..16383
    const int tile = wid & 7;                // 16-wide w tile
    const int head = (wid >> 3) & 63;
    const int b    = wid >> 9;               // 0..31

    const int n     = lane & 15;             // A: M index; B/C: N column
    const int hi    = lane >> 4;             // 0 lower half, 1 upper half
    const bool lower = (hi == 0);
    const int w0    = tile * 16;

    // ---- A matrices (one 16x4 per kw). A[m][k]: m = 4*kh+co (rows 12..15 zero), k = ci.
    // Lane layout (ISA 7.12.2): M = lane&15; VGPR0 holds K={0|2}, VGPR1 holds K={1|3} by half.
    v2f a0, a1, a2;
    {
        const int m = n;
        if (m < 12) {
            const int kh = m >> 2, co = m & 3;
            const int ciA = hi ? 2 : 0;
            const float* wp = wts + (size_t)((head * 4 + co) * 4) * 9 + kh * 3;
            a0.x = wp[ciA * 9 + 0];  a0.y = wp[(ciA + 1) * 9 + 0];
            a1.x = wp[ciA * 9 + 1];  a1.y = wp[(ciA + 1) * 9 + 1];
            a2.x = wp[ciA * 9 + 2];  a2.y = wp[(ciA + 1) * 9 + 2];
        } else {
            a0 = (v2f){0.f, 0.f}; a1 = a0; a2 = a0;
        }
    }

    // Bias (wave-uniform head -> scalar loads)
    const float b0 = bias[head * 4 + 0];
    const float b1 = bias[head * 4 + 1];
    const float b2 = bias[head * 4 + 2];
    const float b3 = bias[head * 4 + 3];

    // Input lane pointers: B[ci][n] = x[b, ci*64+head, h, w0+n+kw-1]; ci = {0,1} lower, {2,3} upper.
    const int ci0 = hi ? 2 : 0;
    const float* x0 = x + (size_t)(b * NCH + (ci0    ) * NHEAD + head) * CHSTR;
    const float* x1 = x + (size_t)(b * NCH + (ci0 + 1) * NHEAD + head) * CHSTR;

    // Store lane pointers (interleaved output: channel = co*64 + head), pairA: co 0|2, pairB: co 1|3.
    const int coA = hi ? 2 : 0;
    const int coB = hi ? 3 : 1;
    float* oA = out + (size_t)(b * NCH + coA * NHEAD + head) * CHSTR + w0 + n;
    float* oB = out + (size_t)(b * NCH + coB * NHEAD + head) * CHSTR + w0 + n;

    if (tile == 0 || tile == 7) {
        // Edge tiles: clamp w and mask out-of-range taps (loop-invariant masks/offsets).
        const int wl = w0 + n - 1;
        int ok[3], cf[3];
#pragma unroll
        for (int kw = 0; kw < 3; ++kw) {
            const int w = wl + kw;
            ok[kw] = ((unsigned)w < (unsigned)HW);
            cf[kw] = w < 0 ? 0 : (w > HW - 1 ? HW - 1 : w);
        }
        conv_sweep<true>(x0, x1, oA, oB, a0, a1, a2, b0, b1, b2, b3, lower,
                         ok[0], ok[1], ok[2], cf[0], cf[1], cf[2]);
    } else {
        // Interior: 2 advancing pointers, immediate offsets 0/1/2 (-4/0/+4 bytes from center).
        conv_sweep<false>(x0 + (w0 + n - 1), x1 + (w0 + n - 1), oA, oB,
                          a0, a1, a2, b0, b1, b2, b3, lower, 0, 0, 0, 0, 1, 2);
    }
}

extern "C" void kernel_launch(void* const* d_in, const int* in_sizes, int n_in,
                              void* d_out, int out_size, void* d_ws, size_t ws_size,
                              hipStream_t stream) {
    const float* x    = (const float*)d_in[0];
    const float* wts  = (const float*)d_in[1];
    const float* bias = (const float*)d_in[2];
    float* out = (float*)d_out;

    // 16384 waves = 32 batch * 64 heads * 8 w-tiles; 8 waves (256 thr) per block.
    dim3 grid(2048), block(256);
    hipLaunchKernelGGL(bdconv_wmma_f32_kernel, grid, block, 0, stream,
                       x, wts, bias, out);
}